// KNN_68977174774512
// MI455X (gfx1250) — compile-verified
//
#include <hip/hip_runtime.h>

// Problem constants (match reference setup_inputs)
#define NQ      2048
#define NT      65536
#define DIM     512
#define KNN     32
#define NCLASS  100

// Tiling
#define QTILE    32                 // queries per block (two WMMA M tiles per wave)
#define CHUNK    1024               // train points per block
#define NCHUNK   (NT / CHUNK)       // 64
#define NQTILE   (NQ / QTILE)       // 64
#define A_STRIDE 516                // 512 + 4 pad: conflict-free A-frag reads, 16B aligned rows
#define BLOCK2   512                // 16 waves

typedef __attribute__((ext_vector_type(2))) float v2f;
typedef __attribute__((ext_vector_type(4))) float v4f;
typedef __attribute__((ext_vector_type(8))) float v8f;

// ---------------------------------------------------------------------------
// Kernel 1: row norms.  One wave (32 lanes) per row; rows [0,NT) = train,
// rows [NT, NT+NQ) = test.
// ---------------------------------------------------------------------------
__global__ void knn_norms(const float* __restrict__ Xt, const float* __restrict__ Xq,
                          float* __restrict__ t2, float* __restrict__ q2) {
    const int wave = (blockIdx.x * blockDim.x + threadIdx.x) >> 5;
    const int lane = threadIdx.x & 31;
    if (wave >= NT + NQ) return;
    const float* row = (wave < NT) ? (Xt + (size_t)wave * DIM)
                                   : (Xq + (size_t)(wave - NT) * DIM);
    float s = 0.f;
    #pragma unroll
    for (int k = lane * 4; k < DIM; k += 32 * 4) {
        v4f v = *(const v4f*)(row + k);
        s += v.x * v.x + v.y * v.y + v.z * v.z + v.w * v.w;
    }
    #pragma unroll
    for (int off = 16; off > 0; off >>= 1) s += __shfl_down(s, off, 32);
    if (lane == 0) {
        if (wave < NT) t2[wave] = s; else q2[wave - NT] = s;
    }
}

// ---------------------------------------------------------------------------
// Kernel 2: fused WMMA GEMM (f32 16x16x4, 2 M-tiles/wave, pipelined B) +
// per-chunk exact top-32.
// Block = 512 threads (16 waves).  blockIdx.x = query tile, blockIdx.y = chunk.
// LDS: A tile [32][516] f32 (66 KB) + distance buffer [32][1024] f32 (128 KB).
// ---------------------------------------------------------------------------
__global__ void knn_chunk(const float* __restrict__ Xq, const float* __restrict__ Xt,
                          const float* __restrict__ q2, const float* __restrict__ t2,
                          float* __restrict__ candD, int* __restrict__ candI) {
    extern __shared__ float smem[];
    float* As = smem;                       // [QTILE][A_STRIDE]
    float* Ds = smem + QTILE * A_STRIDE;    // [QTILE][CHUNK]

    const int tid  = threadIdx.x;
    const int lane = tid & 31;
    const int wv   = tid >> 5;              // wave id 0..15
    const int q0   = blockIdx.x * QTILE;
    const int j0b  = blockIdx.y * CHUNK;

    // ---- Stage query tile (32 x 512 f32) into LDS, coalesced float4 ----
    for (int idx = tid * 4; idx < QTILE * DIM; idx += BLOCK2 * 4) {
        const int r = idx / DIM;
        const int c = idx - r * DIM;        // multiple of 4, DIM % 4 == 0
        v4f v = *(const v4f*)(Xq + (size_t)(q0 + r) * DIM + c);
        *(v4f*)(As + r * A_STRIDE + c) = v; // A_STRIDE % 4 == 0 -> aligned
    }
    __syncthreads();

    // ---- WMMA GEMM ----
    // f32 16x16x4 fragment layout (wave32):
    //   A: lane L holds row M=L%16; VGPR0={K, K+2}, VGPR1={K+1, K+3} split by L<16/L>=16
    //   B: lane L holds col N=L%16; same K split (B is 4x16, K-major striping)
    // Each wave computes a 32x16 output slab: two accumulators sharing B frags.
    const int nn    = lane & 15;
    const int khalf = (lane >> 4) * 2;      // 0 for lanes 0-15, 2 for lanes 16-31
    const float* Arow0 = As + nn * A_STRIDE + khalf;           // rows 0..15
    const float* Arow1 = Arow0 + 16 * A_STRIDE;                // rows 16..31

    for (int t = wv; t < CHUNK / 16; t += 16) {                // 4 col-tiles per wave
        const int j0 = j0b + t * 16;
        const float* Bcol = Xt + (size_t)(j0 + nn) * DIM + khalf;

        v8f acc0 = {};
        v8f acc1 = {};

        // Software-pipelined B fragments: keep one 16-dim group in flight.
        v2f bbuf[4];
        #pragma unroll
        for (int i = 0; i < 4; ++i) bbuf[i] = *(const v2f*)(Bcol + 4 * i);

        for (int k0 = 0; k0 < DIM; k0 += 16) {
            const int kn = (k0 + 16 < DIM) ? (k0 + 16) : k0;   // branchless tail
            v2f bnext[4];
            #pragma unroll
            for (int i = 0; i < 4; ++i) bnext[i] = *(const v2f*)(Bcol + kn + 4 * i);

            #pragma unroll
            for (int i = 0; i < 4; ++i) {
                v2f a0 = *(const v2f*)(Arow0 + k0 + 4 * i);
                v2f a1 = *(const v2f*)(Arow1 + k0 + 4 * i);
                acc0 = __builtin_amdgcn_wmma_f32_16x16x4_f32(
                    false, a0, false, bbuf[i], (short)0, acc0, false, false);
                acc1 = __builtin_amdgcn_wmma_f32_16x16x4_f32(
                    false, a1, false, bbuf[i], (short)0, acc1, false, false);
            }
            #pragma unroll
            for (int i = 0; i < 4; ++i) bbuf[i] = bnext[i];
        }

        // Epilogue: d2 = q2[row] + t2[col] - 2*dot  -> LDS distance buffer.
        // C layout: VGPR r -> row M = r + (lane>=16 ? 8 : 0); col N = lane%16.
        const int mhalf = (lane >> 4) * 8;
        const float tn  = t2[j0 + nn];
        #pragma unroll
        for (int r = 0; r < 8; ++r) {
            const int m0 = mhalf + r;
            const int m1 = 16 + mhalf + r;
            Ds[m0 * CHUNK + t * 16 + nn] = q2[q0 + m0] + tn - 2.0f * acc0[r];
            Ds[m1 * CHUNK + t * 16 + nn] = q2[q0 + m1] + tn - 2.0f * acc1[r];
        }
    }
    __syncthreads();

    // ---- Exact per-query top-32 within the chunk (wave-parallel extraction) ----
    const float FINF = __builtin_inff();
    for (int qq = wv * 2; qq < wv * 2 + 2; ++qq) {             // 32 queries / 16 waves
        float* drow = Ds + qq * CHUNK;
        const int gq = q0 + qq;
        float* outD = candD + ((size_t)gq * NCHUNK + blockIdx.y) * KNN;
        int*   outI = candI + ((size_t)gq * NCHUNK + blockIdx.y) * KNN;

        for (int it = 0; it < KNN; ++it) {
            float best = FINF; int bcol = CHUNK;
            for (int cix = lane; cix < CHUNK; cix += 32) {
                float v = drow[cix];
                if (v < best) { best = v; bcol = cix; }
            }
            #pragma unroll
            for (int off = 16; off > 0; off >>= 1) {
                float ov = __shfl_xor(best, off, 32);
                int   oc = __shfl_xor(bcol, off, 32);
                if (ov < best || (ov == best && oc < bcol)) { best = ov; bcol = oc; }
            }
            if (lane == 0) {
                drow[bcol] = FINF;          // consume winner
                outD[it] = best;
                outI[it] = j0b + bcol;
            }
            __builtin_amdgcn_wave_barrier(); // keep LDS mark ordered vs next scan
        }
    }
}

// ---------------------------------------------------------------------------
// Kernel 3: merge NCHUNK*32 candidates per query -> global top-32 -> majority
// vote over 100 classes.  One block (256 threads) per query.
// ---------------------------------------------------------------------------
__global__ void knn_merge_vote(const float* __restrict__ candD, const int* __restrict__ candI,
                               const int* __restrict__ y_train, int* __restrict__ y_pred) {
    __shared__ float sD[NCHUNK * KNN];      // 2048 floats
    __shared__ int   sI[NCHUNK * KNN];
    __shared__ int   counts[NCLASS];
    __shared__ int   winners[KNN];

    const int q = blockIdx.x;
    const int tid = threadIdx.x;

    for (int i = tid; i < NCHUNK * KNN; i += blockDim.x) {
        sD[i] = candD[(size_t)q * NCHUNK * KNN + i];
        sI[i] = candI[(size_t)q * NCHUNK * KNN + i];
    }
    for (int i = tid; i < NCLASS; i += blockDim.x) counts[i] = 0;
    __syncthreads();

    const float FINF = __builtin_inff();
    if (tid < 32) {                          // wave 0 extracts global top-32
        const int lane = tid;
        for (int it = 0; it < KNN; ++it) {
            float best = FINF; int bix = NCHUNK * KNN;
            for (int cix = lane; cix < NCHUNK * KNN; cix += 32) {
                float v = sD[cix];
                if (v < best) { best = v; bix = cix; }
            }
            #pragma unroll
            for (int off = 16; off > 0; off >>= 1) {
                float ov = __shfl_xor(best, off, 32);
                int   oc = __shfl_xor(bix, off, 32);
                if (ov < best || (ov == best && oc < bix)) { best = ov; bix = oc; }
            }
            if (lane == 0) { sD[bix] = FINF; winners[it] = sI[bix]; }
            __builtin_amdgcn_wave_barrier();
        }
    }
    __syncthreads();

    if (tid < KNN) atomicAdd(&counts[y_train[winners[tid]]], 1);
    __syncthreads();

    if (tid == 0) {                          // argmax, first-max tie-break
        int bestc = 0, bestn = counts[0];
        for (int c = 1; c < NCLASS; ++c)
            if (counts[c] > bestn) { bestn = counts[c]; bestc = c; }
        y_pred[q] = bestc;
    }
}

// ---------------------------------------------------------------------------
// Launch
// ---------------------------------------------------------------------------
extern "C" void kernel_launch(void* const* d_in, const int* in_sizes, int n_in,
                              void* d_out, int out_size, void* d_ws, size_t ws_size,
                              hipStream_t stream) {
    (void)in_sizes; (void)n_in; (void)out_size; (void)ws_size;

    const float* Xq  = (const float*)d_in[0];   // [2048, 512]
    const float* Xt  = (const float*)d_in[1];   // [65536, 512]
    const int*   ytr = (const int*)d_in[2];     // [65536]
    int* ypred = (int*)d_out;                   // [2048] int32

    // Workspace layout (~34 MB total)
    float* t2    = (float*)d_ws;                                    // NT
    float* q2    = t2 + NT;                                         // NQ
    float* candD = q2 + NQ;                                         // NQ*NCHUNK*KNN
    int*   candI = (int*)(candD + (size_t)NQ * NCHUNK * KNN);       // NQ*NCHUNK*KNN

    // 1) norms: one wave per row
    {
        const int waves  = NT + NQ;
        const int blocks = (waves + 7) / 8;     // 8 waves / 256-thread block
        knn_norms<<<blocks, 256, 0, stream>>>(Xt, Xq, t2, q2);
    }

    // 2) fused WMMA distance + per-chunk top-32
    {
        const size_t lds = (size_t)(QTILE * A_STRIDE + QTILE * CHUNK) * sizeof(float); // ~193 KB
        hipFuncSetAttribute((const void*)knn_chunk,
                            hipFuncAttributeMaxDynamicSharedMemorySize, (int)lds);
        dim3 grid(NQTILE, NCHUNK);
        knn_chunk<<<grid, BLOCK2, lds, stream>>>(Xq, Xt, q2, t2, candD, candI);
    }

    // 3) merge + majority vote
    knn_merge_vote<<<NQ, 256, 0, stream>>>(candD, candI, ytr, ypred);
}